// sLayer_36189394436362
// MI455X (gfx1250) — compile-verified
//
#include <hip/hip_runtime.h>
#include <hip/hip_bf16.h>
#include <math.h>

// ---------------------------------------------------------------------------
// GAT-style edge softmax:
//   a      = edge_features @ w_attn            (GEMV, done with V_WMMA_F32_16X16X4_F32)
//   alpha  = segment_softmax(a, src, N)        (atomic max + atomic sum in L2)
//   h      = node_features                     (D2D copy)
// Bandwidth-bound: ~1.05 GB traffic -> ~45us at 23.3 TB/s. WMMA used for the
// GEMV because it is the only f32 matrix path; compute is negligible either way.
// ---------------------------------------------------------------------------

typedef __attribute__((ext_vector_type(2))) float v2f;
typedef __attribute__((ext_vector_type(8))) float v8f;

// ---------------------------------------------------------------------------
// Pass 1: a[e] = dot(edge_features[e, 0:64], w[0:64]) via WMMA f32 16x16x4.
// One wave handles a 16-edge group; 16 chained WMMAs cover K=64.
// f32 A-fragment layout (ISA 16x4): lanes 0-15 hold K={4c,4c+1} of row M=lane,
// lanes 16-31 hold K={4c+2,4c+3} of row M=lane-16  -> one aligned float2 load.
// B (4x16) is the same K mapping with every column = w[k], so the B fragment
// is the identical float2 gather from the weight vector (preloaded, 16x v2f).
// D: VGPR r holds M=r (lanes 0-15) / M=r+8 (lanes 16-31), replicated over N.
// ---------------------------------------------------------------------------
__global__ void __launch_bounds__(256)
gemv_wmma_kernel(const float* __restrict__ ef,   // [E,64]
                 const float* __restrict__ w,    // [64]
                 float* __restrict__ a_out,      // [E]
                 long long E, long long numGroups)
{
    const int lane = threadIdx.x & 31;
    const int wave = threadIdx.x >> 5;
    const int mrow = lane & 15;            // row within 16-edge group
    const int koff = (lane >> 4) << 1;     // 0 for lanes 0-15, 2 for lanes 16-31

    // Preload B fragments (w broadcast across all 16 columns): 16 x float2.
    v2f bfrag[16];
#pragma unroll
    for (int c = 0; c < 16; ++c)
        bfrag[c] = *(const v2f*)(w + 4 * c + koff);

    const long long wavesTotal = (long long)gridDim.x * (blockDim.x >> 5);
    long long g = (long long)blockIdx.x * (blockDim.x >> 5) + wave;

    for (; g < numGroups; g += wavesTotal) {
        const long long ebase = g * 16;
        long long erow = ebase + mrow;
        if (erow >= E) erow = E - 1;                    // clamp (E%16==0 normally)
        const float* ap = ef + erow * 64 + koff;

        v8f acc = {};
#pragma unroll
        for (int c = 0; c < 16; ++c) {
            v2f av = *(const v2f*)(ap + 4 * c);         // b64, 8B-aligned
            acc = __builtin_amdgcn_wmma_f32_16x16x4_f32(
                false, av, false, bfrag[c], (short)0, acc, false, false);
        }

        // Every column of D is identical; emit the 16 scalars with
        // two-lane-predicated stores (lane r and lane 16+r per VGPR r).
#pragma unroll
        for (int r = 0; r < 8; ++r) {
            long long row = ebase + ((lane >> 4) << 3) + r;
            if (((lane & 15) == r) && row < E)
                a_out[row] = acc[r];
        }
    }
}

// ---------------------------------------------------------------------------
// Segment softmax passes. m[] kept as monotone uint encoding of float so the
// max reduction is a plain global_atomic_max_u32 (L2-resident, 400KB table).
// ---------------------------------------------------------------------------
__device__ __forceinline__ unsigned enc_f32(float f) {
    unsigned u = __float_as_uint(f);
    return (u >> 31) ? ~u : (u | 0x80000000u);
}
__device__ __forceinline__ float dec_f32(unsigned e) {
    return (e & 0x80000000u) ? __uint_as_float(e ^ 0x80000000u)
                             : __uint_as_float(~e);
}

__global__ void init_kernel(unsigned* __restrict__ m_u, float* __restrict__ s, int N) {
    int i = blockIdx.x * blockDim.x + threadIdx.x;
    if (i < N) { m_u[i] = 0u; s[i] = 0.0f; }   // enc 0 < enc(-inf): safe floor
}

__global__ void seg_max_kernel(const float* __restrict__ a,
                               const int* __restrict__ src,
                               unsigned* __restrict__ m_u, long long E) {
    long long i = (long long)blockIdx.x * blockDim.x + threadIdx.x;
    if (i < E) atomicMax(&m_u[src[i]], enc_f32(a[i]));
}

__global__ void seg_expsum_kernel(float* __restrict__ a,              // in: a, out: ex
                                  const int* __restrict__ src,
                                  const unsigned* __restrict__ m_u,
                                  float* __restrict__ s, long long E) {
    long long i = (long long)blockIdx.x * blockDim.x + threadIdx.x;
    if (i < E) {
        int g = src[i];
        float ex = __expf(a[i] - dec_f32(m_u[g]));
        a[i] = ex;
        atomicAdd(&s[g], ex);
    }
}

__global__ void seg_norm_kernel(const float* __restrict__ ex,
                                const int* __restrict__ src,
                                const float* __restrict__ s,
                                float* __restrict__ alpha, long long E) {
    long long i = (long long)blockIdx.x * blockDim.x + threadIdx.x;
    if (i < E) alpha[i] = ex[i] / s[src[i]];
}

// ---------------------------------------------------------------------------
// inputs: [0] node_features [N,128] f32, [1] edge_features [E,64] f32,
//         [2] src [E] i32, [3] w_attn [64] f32
// d_out:  h [N*128] f32  ++  alpha [E] f32
// ws:     a/ex [E] f32 | m_u [N] u32 | s [N] f32
// ---------------------------------------------------------------------------
extern "C" void kernel_launch(void* const* d_in, const int* in_sizes, int n_in,
                              void* d_out, int out_size, void* d_ws, size_t ws_size,
                              hipStream_t stream) {
    const float* node = (const float*)d_in[0];
    const float* ef   = (const float*)d_in[1];
    const int*   src  = (const int*)d_in[2];
    const float* w    = (const float*)d_in[3];

    const long long N = in_sizes[0] / 128;
    const long long E = in_sizes[2];

    float*    a_ws = (float*)d_ws;
    unsigned* m_u  = (unsigned*)(a_ws + E);
    float*    s_ws = (float*)((char*)m_u + N * sizeof(unsigned));

    float* h_out     = (float*)d_out;           // [N*128]
    float* alpha_out = h_out + N * 128;         // [E]

    // h = node_features (pure passthrough, 51 MB D2D)
    hipMemcpyAsync(h_out, node, (size_t)N * 128 * sizeof(float),
                   hipMemcpyDeviceToDevice, stream);

    // Pass 1: GEMV via WMMA. 16 edges per wave-group, persistent waves.
    {
        long long numGroups = (E + 15) / 16;
        long long wantBlocks = (numGroups + 7) / 8;          // 8 waves/block
        int blocks = (int)(wantBlocks < 4096 ? wantBlocks : 4096);
        if (blocks < 1) blocks = 1;
        gemv_wmma_kernel<<<blocks, 256, 0, stream>>>(ef, w, a_ws, E, numGroups);
    }

    // Pass 2: init segment tables (must run every call: graph replay).
    {
        int blocks = (int)((N + 255) / 256);
        init_kernel<<<blocks, 256, 0, stream>>>(m_u, s_ws, (int)N);
    }

    const int eBlocks = (int)((E + 255) / 256);
    // Pass 3: segment max (atomic u32 max on L2-resident table).
    seg_max_kernel<<<eBlocks, 256, 0, stream>>>(a_ws, src, m_u, E);
    // Pass 4: ex = exp(a - m[src]) in place; segment sum via atomic f32 add.
    seg_expsum_kernel<<<eBlocks, 256, 0, stream>>>(a_ws, src, m_u, s_ws, E);
    // Pass 5: alpha = ex / s[src].
    seg_norm_kernel<<<eBlocks, 256, 0, stream>>>(a_ws, src, s_ws, alpha_out, E);
}